// RoutingFunction_18442589569252
// MI455X (gfx1250) — compile-verified
//
#include <hip/hip_runtime.h>
#include <math.h>

typedef __attribute__((ext_vector_type(2))) float v2f;
typedef __attribute__((ext_vector_type(4))) float v4f;
typedef __attribute__((ext_vector_type(8))) float v8f;

#define Bn 128
#define Cn 512
#define HWn 1024
#define En 16
#define EPSf 1e-8f

// ---------------- Kernel 1: adaptive average pool (HBM-bound, 256 MB stream) ----
// One wave32 per (b,c) plane of 1024 contiguous floats. 8 x float4 per lane,
// non-temporal (read-once, > L2 capacity), then wave shuffle reduction.
__global__ void moe_pool_kernel(const float* __restrict__ x,
                                float* __restrict__ pooled) {
    int wave = (blockIdx.x * blockDim.x + threadIdx.x) >> 5; // (b,c) plane id
    int lane = threadIdx.x & 31;
    const v4f* p = (const v4f*)(x + (size_t)wave * HWn);
    float s = 0.0f;
#pragma unroll
    for (int w = 0; w < 8; ++w) {
        v4f v = __builtin_nontemporal_load(&p[w * 32 + lane]);
        s += v.x + v.y + v.z + v.w;
    }
#pragma unroll
    for (int off = 16; off > 0; off >>= 1)
        s += __shfl_xor(s, off, 32);
    if (lane == 0)
        pooled[wave] = s * (1.0f / 1024.0f);
}

// ---------------- Kernel 2: logits = pooled @ W_gate^T via V_WMMA_F32_16X16X4_F32
// D[128,16] = A[128,512] x B[512,16], B[k][n] = W_gate[n][k].
// One block, 8 waves; wave w owns M-tile rows [16w, 16w+16). 128 K-steps of 4.
__global__ void moe_gate_gemm_kernel(const float* __restrict__ pooled,
                                     const float* __restrict__ Wg,
                                     float* __restrict__ logits) {
    int wave = threadIdx.x >> 5;   // 0..7 -> M tile
    int lane = threadIdx.x & 31;
    int half = lane >> 4;          // selects K pair {0,1} vs {2,3}
    int l    = lane & 15;          // row (A) / col (B) within tile

    const float* arow = pooled + (size_t)(wave * 16 + l) * Cn + half * 2;
    const float* brow = Wg     + (size_t)l * Cn + half * 2;

    v8f c = {};
    for (int k0 = 0; k0 < Cn; k0 += 4) {
        v2f a = *(const v2f*)(arow + k0);
        v2f b = *(const v2f*)(brow + k0);
        // (neg_a, A, neg_b, B, c_mod, C, reuse_a, reuse_b)
        c = __builtin_amdgcn_wmma_f32_16x16x4_f32(false, a, false, b,
                                                  (short)0, c, false, false);
    }
    // C/D layout: VGPR v -> row m0 + half*8 + v, col = l
    int rbase = wave * 16 + half * 8;
#pragma unroll
    for (int v = 0; v < 8; ++v)
        logits[(size_t)(rbase + v) * En + l] = c[v];
}

// ---------------- Kernel 3: softmax / top-2 / aux losses (tiny, one block) -----
__global__ void moe_finalize_kernel(const float* __restrict__ logits,
                                    const float* __restrict__ complexity,
                                    const float* __restrict__ noise,
                                    float* __restrict__ out) {
    __shared__ float smat[Bn][En];   // softmax(logits) rows
    __shared__ float pmat[Bn][En];   // p rows for load loss
    __shared__ float impA[En];
    __shared__ float pmA[En];
    __shared__ float lossA[2];

    int b = threadIdx.x;             // 0..127, one row per thread

    float lg[En], s[En], nl[En], g[En];
#pragma unroll
    for (int e = 0; e < En; ++e) lg[e] = logits[b * En + e];

    // softmax of clean logits -> importance
    float m = lg[0];
#pragma unroll
    for (int e = 1; e < En; ++e) m = fmaxf(m, lg[e]);
    float sum = 0.0f;
#pragma unroll
    for (int e = 0; e < En; ++e) { s[e] = __expf(lg[e] - m); sum += s[e]; }
    float inv = 1.0f / sum;
#pragma unroll
    for (int e = 0; e < En; ++e) smat[b][e] = s[e] * inv;

    // noisy logits + their softmax (gating scores)
#pragma unroll
    for (int e = 0; e < En; ++e) nl[e] = lg[e] + noise[b * En + e];
    float m2 = nl[0];
#pragma unroll
    for (int e = 1; e < En; ++e) m2 = fmaxf(m2, nl[e]);
    float sum2 = 0.0f;
#pragma unroll
    for (int e = 0; e < En; ++e) { g[e] = __expf(nl[e] - m2); sum2 += g[e]; }
    float inv2 = 1.0f / sum2;

    // top-2 of noisy logits (same indices as top-2 of gating scores;
    // strict '>' keeps lowest index on ties, matching jax top_k)
    int i0 = 0; float v0 = nl[0];
#pragma unroll
    for (int e = 1; e < En; ++e) if (nl[e] > v0) { v0 = nl[e]; i0 = e; }
    int i1 = -1; float v1 = -3.4e38f;
#pragma unroll
    for (int e = 0; e < En; ++e)
        if (e != i0 && nl[e] > v1) { v1 = nl[e]; i1 = e; }
    float thr = v1;                        // k-th (2nd) largest noisy logit
    float g0 = g[i0] * inv2, g1 = g[i1] * inv2;

    // dense gates + indices + values
#pragma unroll
    for (int e = 0; e < En; ++e)
        out[b * En + e] = (e == i0) ? g0 : ((e == i1) ? g1 : 0.0f);
    out[2048 + b * 2 + 0] = (float)i0;
    out[2048 + b * 2 + 1] = (float)i1;
    out[2304 + b * 2 + 0] = g0;
    out[2304 + b * 2 + 1] = g1;

    // p = 1 - Phi((thr - logits)/NOISE_STD), NOISE_STD = 1/16
    const float inv_sqrt2 = 0.70710678118654752440f;
#pragma unroll
    for (int e = 0; e < En; ++e) {
        float z = (thr - lg[e]) * 16.0f;
        pmat[b][e] = 1.0f - 0.5f * (1.0f + erff(z * inv_sqrt2));
    }
    __syncthreads();

    // deterministic column reductions over B (threads 0..15)
    if (b < En) {
        float si = 0.0f, sp = 0.0f;
        for (int r = 0; r < Bn; ++r) { si += smat[r][b]; sp += pmat[r][b]; }
        impA[b] = si * complexity[b];          // TAU == 1
        pmA[b]  = sp * (1.0f / (float)Bn);
    }
    __syncthreads();

    if (b == 0) {
        float mi = 0.0f, mp = 0.0f;
#pragma unroll
        for (int e = 0; e < En; ++e) { mi += impA[e]; mp += pmA[e]; }
        mi *= (1.0f / En); mp *= (1.0f / En);
        float vi = 0.0f, vp = 0.0f;
#pragma unroll
        for (int e = 0; e < En; ++e) {
            float di = impA[e] - mi; vi += di * di;
            float dp = pmA[e] - mp;  vp += dp * dp;
        }
        vi *= (1.0f / (En - 1)); vp *= (1.0f / (En - 1));   // ddof=1
        float li = sqrtf(vi) / (mi + EPSf);
        float ll = sqrtf(vp) / (mp + EPSf);
        lossA[0] = 0.5f * li * li + 0.5f * ll * ll;
        out[2560] = lossA[0];
    }
}

extern "C" void kernel_launch(void* const* d_in, const int* in_sizes, int n_in,
                              void* d_out, int out_size, void* d_ws, size_t ws_size,
                              hipStream_t stream) {
    const float* x     = (const float*)d_in[0];   // [128,512,32,32]
    const float* Wg    = (const float*)d_in[1];   // [16,512]
    const float* comp  = (const float*)d_in[2];   // [16]
    const float* noise = (const float*)d_in[3];   // [128,16]
    float* out = (float*)d_out;                   // 2048 + 256 + 256 + 1

    float* pooled = (float*)d_ws;                 // [128,512] = 256 KB
    float* logits = pooled + Bn * Cn;             // [128,16]

    // 65536 (b,c) planes, one wave each, 8 waves per 256-thread block
    moe_pool_kernel<<<dim3((Bn * Cn) / 8), 256, 0, stream>>>(x, pooled);
    moe_gate_gemm_kernel<<<1, 256, 0, stream>>>(pooled, Wg, logits);
    moe_finalize_kernel<<<1, Bn, 0, stream>>>(logits, comp, noise, out);
}